// GCN_3848290697713
// MI455X (gfx1250) — compile-verified
//
#include <hip/hip_runtime.h>

// GCN, 2 layers: out = norm_agg(relu(norm_agg(X@W1)+b1) @ W2) + b2
// N=50000 nodes, E=1.6M edges, IN=256, HID=128, OUT=40.
// GEMMs via V_WMMA_F32_16X16X4_F32 (fp32 — GEMM is not the bottleneck, keep
// reference precision). Aggregation via L2-resident gather + f32 atomics.

typedef __attribute__((ext_vector_type(2))) float v2f;
typedef __attribute__((ext_vector_type(8))) float v8f;

#define N_NODES 50000
#define N_EDGES 1600000
#define F_IN    256
#define F_HID   128
#define F_OUT   40

// ---------------- degree / normalization ----------------
__global__ void k_init_deg(float* __restrict__ deg) {
  int i = blockIdx.x * blockDim.x + threadIdx.x;
  if (i < N_NODES) deg[i] = 1.0f;            // self-loop contribution
}

__global__ void k_count_deg(const int* __restrict__ ei, float* __restrict__ deg) {
  int e = blockIdx.x * blockDim.x + threadIdx.x;
  if (e < N_EDGES) atomicAdd(&deg[ei[N_EDGES + e]], 1.0f);   // dst = ei[1][e]
}

__global__ void k_deg_to_dinv(float* __restrict__ deg) {
  int i = blockIdx.x * blockDim.x + threadIdx.x;
  if (i < N_NODES) {
    float d = deg[i];
    deg[i] = (d > 0.0f) ? rsqrtf(d) : 0.0f;
  }
}

// ---------------- GEMM1: H1[N,128] = X[N,256] @ W1[256,128] ----------------
// One wave -> 16 rows x 128 cols (8 accumulators), K loop of 64 WMMA steps.
__global__ void __launch_bounds__(128)
k_gemm1_wmma(const float* __restrict__ X, const float* __restrict__ W,
             float* __restrict__ H) {
  const int lane  = threadIdx.x & 31;
  const int mtile = blockIdx.x * 4 + (threadIdx.x >> 5);
  if (mtile * 16 >= N_NODES) return;          // wave-uniform exit, EXEC stays full
  const int lrow = lane & 15;                 // M (for A) / N (for B,C,D) within tile
  const int kgrp = (lane >> 4) << 1;          // 0 or 2: K group per half-wave

  const float* xrow = X + (size_t)(mtile * 16 + lrow) * F_IN;
  v8f acc[8];
#pragma unroll
  for (int n = 0; n < 8; ++n) acc[n] = (v8f){0,0,0,0,0,0,0,0};

  for (int k = 0; k < F_IN; k += 4) {
    v2f a;                                    // A 16x4: lane holds M=lrow, K=kgrp..kgrp+1
    a.x = xrow[k + kgrp];
    a.y = xrow[k + kgrp + 1];
#pragma unroll
    for (int n = 0; n < 8; ++n) {
      const float* wc = W + (size_t)(k + kgrp) * F_HID + n * 16 + lrow;
      v2f b;                                  // B 4x16: VGPR v -> K=kgrp+v, N=lrow
      b.x = wc[0];
      b.y = wc[F_HID];
      acc[n] = __builtin_amdgcn_wmma_f32_16x16x4_f32(
          false, a, false, b, (short)0, acc[n], false, false);
    }
  }
  // C/D layout: VGPR v -> row = v + 8*(lane>>4), col = lrow
  const int rbase = mtile * 16 + ((lane >> 4) << 3);
#pragma unroll
  for (int n = 0; n < 8; ++n)
#pragma unroll
    for (int v = 0; v < 8; ++v)
      H[(size_t)(rbase + v) * F_HID + n * 16 + lrow] = acc[n][v];
}

// ---------------- GEMM2: G[N,40] = A1[N,128] @ W2[128,40] ----------------
// N padded to 48 (3 tiles), columns >= 40 masked to zero.
__global__ void __launch_bounds__(128)
k_gemm2_wmma(const float* __restrict__ Hin, const float* __restrict__ W,
             float* __restrict__ G) {
  const int lane  = threadIdx.x & 31;
  const int mtile = blockIdx.x * 4 + (threadIdx.x >> 5);
  if (mtile * 16 >= N_NODES) return;
  const int lrow = lane & 15;
  const int kgrp = (lane >> 4) << 1;

  const float* hrow = Hin + (size_t)(mtile * 16 + lrow) * F_HID;
  v8f acc[3];
#pragma unroll
  for (int n = 0; n < 3; ++n) acc[n] = (v8f){0,0,0,0,0,0,0,0};

  for (int k = 0; k < F_HID; k += 4) {
    v2f a;
    a.x = hrow[k + kgrp];
    a.y = hrow[k + kgrp + 1];
#pragma unroll
    for (int n = 0; n < 3; ++n) {
      const int col = n * 16 + lrow;
      v2f b;
      b.x = (col < F_OUT) ? W[(size_t)(k + kgrp) * F_OUT + col]     : 0.0f;
      b.y = (col < F_OUT) ? W[(size_t)(k + kgrp + 1) * F_OUT + col] : 0.0f;
      acc[n] = __builtin_amdgcn_wmma_f32_16x16x4_f32(
          false, a, false, b, (short)0, acc[n], false, false);
    }
  }
  const int rbase = mtile * 16 + ((lane >> 4) << 3);
#pragma unroll
  for (int n = 0; n < 3; ++n) {
    const int col = n * 16 + lrow;
    if (col < F_OUT) {                        // divergence only after all WMMAs
#pragma unroll
      for (int v = 0; v < 8; ++v)
        G[(size_t)(rbase + v) * F_OUT + col] = acc[n][v];
    }
  }
}

// ---------------- aggregation layer 1 (128 features) ----------------
// Init accumulator with the self-loop term: A[i] = H[i] * dinv[i]^2
__global__ void k_selfloop1(const float* __restrict__ H, const float* __restrict__ dinv,
                            float* __restrict__ A) {
  int t = blockIdx.x * blockDim.x + threadIdx.x;
  if (t < N_NODES * F_HID) {
    float d = dinv[t >> 7];
    A[t] = H[t] * d * d;
  }
}

// One wave per edge; lane handles one float4 of the 128-float row.
__global__ void k_agg1(const int* __restrict__ ei, const float* __restrict__ H,
                       const float* __restrict__ dinv, float* __restrict__ A) {
  int t = blockIdx.x * blockDim.x + threadIdx.x;
  int e = t >> 5;
  if (e >= N_EDGES) return;
  int lane = t & 31;
  int src = ei[e];
  int dst = ei[N_EDGES + e];
  float nrm = dinv[src] * dinv[dst];
  float4 h = reinterpret_cast<const float4*>(H + (size_t)src * F_HID)[lane];
  float* out = A + (size_t)dst * F_HID + lane * 4;
  atomicAdd(out + 0, h.x * nrm);
  atomicAdd(out + 1, h.y * nrm);
  atomicAdd(out + 2, h.z * nrm);
  atomicAdd(out + 3, h.w * nrm);
}

__global__ void k_relu_bias(float* __restrict__ A, const float* __restrict__ b1) {
  int t = blockIdx.x * blockDim.x + threadIdx.x;
  if (t < N_NODES * F_HID)
    A[t] = fmaxf(A[t] + b1[t & (F_HID - 1)], 0.0f);
}

// ---------------- aggregation layer 2 (40 features) ----------------
__global__ void k_selfloop2(const float* __restrict__ G, const float* __restrict__ dinv,
                            const float* __restrict__ b2, float* __restrict__ out) {
  int t = blockIdx.x * blockDim.x + threadIdx.x;
  if (t < N_NODES * F_OUT) {
    int i = t / F_OUT;
    int f = t - i * F_OUT;
    float d = dinv[i];
    out[t] = G[t] * d * d + b2[f];
  }
}

__global__ void k_agg2(const int* __restrict__ ei, const float* __restrict__ G,
                       const float* __restrict__ dinv, float* __restrict__ out) {
  int t = blockIdx.x * blockDim.x + threadIdx.x;
  if (t >= N_EDGES * F_OUT) return;
  int e = t / F_OUT;
  int f = t - e * F_OUT;
  int src = ei[e];
  int dst = ei[N_EDGES + e];
  float nrm = dinv[src] * dinv[dst];
  atomicAdd(&out[(size_t)dst * F_OUT + f], G[(size_t)src * F_OUT + f] * nrm);
}

// ---------------- launch ----------------
extern "C" void kernel_launch(void* const* d_in, const int* in_sizes, int n_in,
                              void* d_out, int out_size, void* d_ws, size_t ws_size,
                              hipStream_t stream) {
  (void)in_sizes; (void)n_in; (void)out_size; (void)ws_size;
  const float* x  = (const float*)d_in[0];
  const int*   ei = (const int*)  d_in[1];   // [2,E]: row 0 = src, row 1 = dst
  const float* W1 = (const float*)d_in[2];
  const float* b1 = (const float*)d_in[3];
  const float* W2 = (const float*)d_in[4];
  const float* b2 = (const float*)d_in[5];
  float* out = (float*)d_out;

  // workspace layout (floats), 16B-aligned slices; total ~59.4 MB
  float* ws   = (float*)d_ws;
  float* dinv = ws;                                  // N
  float* H1   = ws + 50048;                          // N*128
  float* A1   = H1 + (size_t)N_NODES * F_HID;        // N*128
  float* G    = A1 + (size_t)N_NODES * F_HID;        // N*40

  const int T = 256;
  const int nodeBlks  = (N_NODES + T - 1) / T;
  const int edgeBlks  = (N_EDGES + T - 1) / T;
  const int hidBlks   = (N_NODES * F_HID + T - 1) / T;
  const int outBlks   = (N_NODES * F_OUT + T - 1) / T;
  const int gemmBlks  = (N_NODES / 16 + 3) / 4;      // 3125 m-tiles, 4 waves/block
  const int agg1Blks  = (N_EDGES * 32 + T - 1) / T;
  const int agg2Blks  = (N_EDGES * F_OUT + T - 1) / T;

  k_init_deg  <<<nodeBlks, T, 0, stream>>>(dinv);
  k_count_deg <<<edgeBlks, T, 0, stream>>>(ei, dinv);
  k_deg_to_dinv<<<nodeBlks, T, 0, stream>>>(dinv);

  k_gemm1_wmma<<<gemmBlks, 128, 0, stream>>>(x, W1, H1);
  k_selfloop1 <<<hidBlks,  T, 0, stream>>>(H1, dinv, A1);
  k_agg1      <<<agg1Blks, T, 0, stream>>>(ei, H1, dinv, A1);
  k_relu_bias <<<hidBlks,  T, 0, stream>>>(A1, b1);

  k_gemm2_wmma<<<gemmBlks, 128, 0, stream>>>(A1, W2, G);
  k_selfloop2 <<<outBlks,  T, 0, stream>>>(G, dinv, b2, out);
  k_agg2      <<<agg2Blks, T, 0, stream>>>(ei, G, dinv, out);
}